// GloAttnConv_90091234001380
// MI455X (gfx1250) — compile-verified
//
#include <hip/hip_runtime.h>
#include <hip/hip_bf16.h>

#define CDIM 128

typedef float v2f __attribute__((ext_vector_type(2)));
typedef float v8f __attribute__((ext_vector_type(8)));

static __device__ __forceinline__ v8f wmma4(v2f a, v2f b, v8f c) {
  // D = A(16x4 f32) * B(4x16 f32) + C(16x16 f32)
  return __builtin_amdgcn_wmma_f32_16x16x4_f32(false, a, false, b, (short)0, c,
                                               false, false);
}

// ---------------- utility kernels ----------------

__global__ void k_zero_f32(float* p, long long n) {
  long long i = (long long)blockIdx.x * blockDim.x + threadIdx.x;
  long long stride = (long long)gridDim.x * blockDim.x;
  for (; i < n; i += stride) p[i] = 0.0f;
}

__global__ void k_init_cur_acc(const float* __restrict__ x, float* __restrict__ cur,
                               float* __restrict__ acc, long long n) {
  long long i = (long long)blockIdx.x * blockDim.x + threadIdx.x;
  if (i < n) { float v = x[i]; cur[i] = v; acc[i] = v; }
}

__global__ void k_scan_offsets(const int* __restrict__ nn, int* __restrict__ offs, int B) {
  if (blockIdx.x == 0 && threadIdx.x == 0) {
    int s = 0; offs[0] = 0;
    for (int b = 0; b < B; ++b) { s += nn[b]; offs[b + 1] = s; }
  }
}

__global__ void k_assign_batch(const int* __restrict__ offs, int* __restrict__ batch,
                               int N, int B) {
  int n = blockIdx.x * blockDim.x + threadIdx.x;
  if (n >= N) return;
  int lo = 0, hi = B;
  while (hi - lo > 1) { int mid = (lo + hi) >> 1; if (offs[mid] <= n) lo = mid; else hi = mid; }
  batch[n] = lo;
}

__global__ void k_deg(const int* __restrict__ row, const float* __restrict__ w,
                      float* __restrict__ deg, int E) {
  int e = blockIdx.x * blockDim.x + threadIdx.x;
  if (e < E) atomicAdd(&deg[row[e]], w[e]);
}

__global__ void k_dinv(float* deg, int N) {
  int n = blockIdx.x * blockDim.x + threadIdx.x;
  if (n < N) { float d = deg[n]; deg[n] = (d > 0.0f) ? rsqrtf(d) : 0.0f; }
}

// one wave per edge; lane handles 4 channels (float4): gcn[row] += cur[col]*coef
__global__ void k_edge_scatter(const int* __restrict__ row, const int* __restrict__ col,
                               const float* __restrict__ w, const float* __restrict__ dinv,
                               const float* __restrict__ cur, float* __restrict__ gcn, int E) {
  int gid = blockIdx.x * blockDim.x + threadIdx.x;
  int e = gid >> 5;
  int lane = gid & 31;
  if (e >= E) return;
  int r = row[e], c = col[e];
  float coef = dinv[r] * dinv[c] * w[e];
  const float4* src = (const float4*)(cur + (long long)c * CDIM);
  float4 v = src[lane];
  float* dst = gcn + (long long)r * CDIM + lane * 4;
  atomicAdd(dst + 0, v.x * coef);
  atomicAdd(dst + 1, v.y * coef);
  atomicAdd(dst + 2, v.z * coef);
  atomicAdd(dst + 3, v.w * coef);
}

// per graph: z[b][c] = sum_n ks[n][c]; vsum[b][c] = sum_n cur[n][c]
__global__ void k_sumkv(const float* __restrict__ ks, const float* __restrict__ cur,
                        const int* __restrict__ offs, float* __restrict__ z,
                        float* __restrict__ vsum) {
  int b = blockIdx.x;
  int c = threadIdx.x;   // 128 threads
  int n0 = offs[b], n1 = offs[b + 1];
  float zz = 0.0f, vv = 0.0f;
  for (int n = n0; n < n1; ++n) {
    zz += ks[(long long)n * CDIM + c];
    vv += cur[(long long)n * CDIM + c];
  }
  z[b * CDIM + c] = zz;
  vsum[b * CDIM + c] = vv;
}

// denom[n] = dot(qs[n], z[batch[n]]) + n_nodes[batch[n]]
__global__ void k_denom(const float* __restrict__ qs, const float* __restrict__ z,
                        const int* __restrict__ batch, const int* __restrict__ nn,
                        float* __restrict__ denom, int N) {
  int n = blockIdx.x * blockDim.x + threadIdx.x;
  if (n >= N) return;
  int b = batch[n];
  const float4* q = (const float4*)(qs + (long long)n * CDIM);
  const float4* zb = (const float4*)(z + b * CDIM);
  float d = (float)nn[b];
#pragma unroll
  for (int i = 0; i < CDIM / 4; ++i) {
    float4 a = q[i], c4 = zb[i];
    d += a.x * c4.x + a.y * c4.y + a.z * c4.z + a.w * c4.w;
  }
  denom[n] = d;
}

// ---------------- WMMA kernels ----------------
// Paired LDS layout: element (k, col) stored at smem[(k>>1)*(2*CDIM) + col*2 + (k&1)].
// A b64 read of the (k,k+1) pair for one col is 8B-aligned; lane stride is 2 floats
// across 16 lanes -> 32 distinct banks, conflict-free.

// Y[n][o] = sum_c X[n][c] * W[o][c] (+ bias[o]); optional row L2-normalize.
// 256 threads = 8 waves; each wave owns a 16-row strip, all 128 output columns.
__global__ __launch_bounds__(256) void k_wmma_gemm_nt(
    const float* __restrict__ X, const float* __restrict__ W,
    const float* __restrict__ bias, float* __restrict__ Y, int N, int normalize) {
  extern __shared__ float sW2[];  // 64 KB dynamic LDS, paired on c

  for (int idx = threadIdx.x; idx < CDIM * CDIM / 4; idx += 256) {
    int o = idx >> 5;
    int c0 = (idx & 31) * 4;
    float4 w4 = ((const float4*)(W + (long long)o * CDIM))[idx & 31];
    float wv[4] = {w4.x, w4.y, w4.z, w4.w};
#pragma unroll
    for (int t = 0; t < 4; ++t) {
      int c = c0 + t;
      sW2[(c >> 1) * (2 * CDIM) + o * 2 + (c & 1)] = wv[t];
    }
  }
  __syncthreads();

  int wave = blockIdx.x * 8 + (threadIdx.x >> 5);
  int r0 = wave * 16;
  if (r0 >= N) return;
  int lane = threadIdx.x & 31;
  int half = lane >> 4, l16 = lane & 15;

  v8f accv[8];
  v8f zero = {0.f, 0.f, 0.f, 0.f, 0.f, 0.f, 0.f, 0.f};
#pragma unroll
  for (int j = 0; j < 8; ++j) accv[j] = zero;

  const float* xrow = X + (long long)(r0 + l16) * CDIM;
#pragma unroll
  for (int k = 0; k < CDIM; k += 4) {
    int ka = k + 2 * half;  // even
    v2f a; a.x = xrow[ka]; a.y = xrow[ka + 1];
    v2f bf[8];
#pragma unroll
    for (int j = 0; j < 8; ++j)
      bf[j] = *(const v2f*)(sW2 + (ka >> 1) * (2 * CDIM) + (j * 16 + l16) * 2);
#pragma unroll
    for (int j = 0; j < 8; ++j) accv[j] = wmma4(a, bf[j], accv[j]);
  }

#pragma unroll
  for (int j = 0; j < 8; ++j) {
    float bj = bias[j * 16 + l16];
#pragma unroll
    for (int v = 0; v < 8; ++v) accv[j][v] += bj;
  }

  if (normalize) {
    float ss[8];
#pragma unroll
    for (int v = 0; v < 8; ++v) {
      float s = 0.f;
#pragma unroll
      for (int j = 0; j < 8; ++j) s += accv[j][v] * accv[j][v];
      s += __shfl_xor(s, 1, 32);
      s += __shfl_xor(s, 2, 32);
      s += __shfl_xor(s, 4, 32);
      s += __shfl_xor(s, 8, 32);
      ss[v] = rsqrtf(s);
    }
#pragma unroll
    for (int j = 0; j < 8; ++j)
#pragma unroll
      for (int v = 0; v < 8; ++v) accv[j][v] *= ss[v];
  }

#pragma unroll
  for (int v = 0; v < 8; ++v) {
    long long rowg = r0 + v + 8 * half;
#pragma unroll
    for (int j = 0; j < 8; ++j) Y[rowg * CDIM + j * 16 + l16] = accv[j][v];
  }
}

// S_b[i][j] = sum_{n in graph b} ks[n][i] * cur[n][j]   (K^T V per graph).
// grid = B*8 blocks, 256 threads (8 waves); block owns row-tile ti=blockIdx&7,
// wave owns col-tile tj. 32-node chunks of ks/cur staged in LDS, paired on node.
__global__ __launch_bounds__(256) void k_wmma_ktv(
    const float* __restrict__ ks, const float* __restrict__ cur,
    const int* __restrict__ offs, float* __restrict__ S) {
  __shared__ float sK2[32 * CDIM];  // 16 KB
  __shared__ float sV2[32 * CDIM];  // 16 KB
  int b = blockIdx.x >> 3;
  int ti = blockIdx.x & 7;
  int tj = threadIdx.x >> 5;
  int lane = threadIdx.x & 31;
  int half = lane >> 4, l16 = lane & 15;
  int n0 = offs[b], n1 = offs[b + 1];

  v8f acc = {0.f, 0.f, 0.f, 0.f, 0.f, 0.f, 0.f, 0.f};
  for (int base = n0; base < n1; base += 32) {
    for (int idx = threadIdx.x; idx < 32 * 32; idx += 256) {
      int r = idx >> 5;
      int c0 = (idx & 31) * 4;
      int node = base + r;
      float4 kv = {0.f, 0.f, 0.f, 0.f}, vv = {0.f, 0.f, 0.f, 0.f};
      if (node < n1) {
        kv = ((const float4*)(ks + (long long)node * CDIM))[idx & 31];
        vv = ((const float4*)(cur + (long long)node * CDIM))[idx & 31];
      }
      float ka4[4] = {kv.x, kv.y, kv.z, kv.w};
      float va4[4] = {vv.x, vv.y, vv.z, vv.w};
#pragma unroll
      for (int t = 0; t < 4; ++t) {
        int c = c0 + t;
        sK2[(r >> 1) * (2 * CDIM) + c * 2 + (r & 1)] = ka4[t];
        sV2[(r >> 1) * (2 * CDIM) + c * 2 + (r & 1)] = va4[t];
      }
    }
    __syncthreads();
#pragma unroll
    for (int kk = 0; kk < 32; kk += 4) {
      int kl = kk + 2 * half;  // even
      v2f a = *(const v2f*)(sK2 + (kl >> 1) * (2 * CDIM) + (ti * 16 + l16) * 2);
      v2f bb = *(const v2f*)(sV2 + (kl >> 1) * (2 * CDIM) + (tj * 16 + l16) * 2);
      acc = wmma4(a, bb, acc);
    }
    __syncthreads();
  }
  float* Sb = S + (long long)b * CDIM * CDIM;
#pragma unroll
  for (int v = 0; v < 8; ++v)
    Sb[(long long)(ti * 16 + v + 8 * half) * CDIM + tj * 16 + l16] = acc[v];
}

// attn numerator via Q_b @ S_b, fused epilogue:
//   attn = (numer + vsum_b) / denom;  cur = 0.5*gcn + 0.5*attn;  acc += cur
__global__ __launch_bounds__(256) void k_wmma_attn_fuse(
    const float* __restrict__ qs, const float* __restrict__ S,
    const float* __restrict__ vsum, const float* __restrict__ denom,
    const float* __restrict__ gcn, const int* __restrict__ batch,
    float* __restrict__ cur, float* __restrict__ accbuf, int N) {
  extern __shared__ float sS2[];  // 64 KB dynamic LDS, paired on i

  int rfirst = blockIdx.x * 128;
  int rlast = rfirst + 127;
  if (rlast >= N) rlast = N - 1;
  int b_first = batch[rfirst];
  bool sameb = (batch[rlast] == b_first);
  if (sameb) {
    const float* Sg = S + (long long)b_first * CDIM * CDIM;
    for (int idx = threadIdx.x; idx < CDIM * CDIM / 4; idx += 256) {
      int i = idx >> 5;
      int j0 = (idx & 31) * 4;
      float4 s4 = ((const float4*)(Sg + (long long)i * CDIM))[idx & 31];
      float sv[4] = {s4.x, s4.y, s4.z, s4.w};
#pragma unroll
      for (int t = 0; t < 4; ++t)
        sS2[(i >> 1) * (2 * CDIM) + (j0 + t) * 2 + (i & 1)] = sv[t];
    }
  }
  __syncthreads();

  int wave = blockIdx.x * 8 + (threadIdx.x >> 5);
  int r0 = wave * 16;
  if (r0 >= N) return;
  int lane = threadIdx.x & 31;
  int half = lane >> 4, l16 = lane & 15;
  int b = batch[r0];
  const float* Sb = S + (long long)b * CDIM * CDIM;

  v8f accv[8];
  v8f zero = {0.f, 0.f, 0.f, 0.f, 0.f, 0.f, 0.f, 0.f};
#pragma unroll
  for (int j = 0; j < 8; ++j) accv[j] = zero;

  const float* qrow = qs + (long long)(r0 + l16) * CDIM;
  if (sameb) {
#pragma unroll
    for (int k = 0; k < CDIM; k += 4) {
      int ka = k + 2 * half;
      v2f a; a.x = qrow[ka]; a.y = qrow[ka + 1];
      v2f bf[8];
#pragma unroll
      for (int j = 0; j < 8; ++j)
        bf[j] = *(const v2f*)(sS2 + (ka >> 1) * (2 * CDIM) + (j * 16 + l16) * 2);
#pragma unroll
      for (int j = 0; j < 8; ++j) accv[j] = wmma4(a, bf[j], accv[j]);
    }
  } else {
#pragma unroll
    for (int k = 0; k < CDIM; k += 4) {
      int ka = k + 2 * half;
      v2f a; a.x = qrow[ka]; a.y = qrow[ka + 1];
      v2f bf[8];
#pragma unroll
      for (int j = 0; j < 8; ++j) {
        int colc = j * 16 + l16;
        bf[j].x = Sb[(long long)ka * CDIM + colc];
        bf[j].y = Sb[(long long)(ka + 1) * CDIM + colc];
      }
#pragma unroll
      for (int j = 0; j < 8; ++j) accv[j] = wmma4(a, bf[j], accv[j]);
    }
  }

  float dn[8];
#pragma unroll
  for (int v = 0; v < 8; ++v) dn[v] = 1.0f / denom[r0 + v + 8 * half];
  float vs[8];
#pragma unroll
  for (int j = 0; j < 8; ++j) vs[j] = vsum[b * CDIM + j * 16 + l16];

#pragma unroll
  for (int v = 0; v < 8; ++v) {
    long long rowg = r0 + v + 8 * half;
#pragma unroll
    for (int j = 0; j < 8; ++j) {
      long long idx = rowg * CDIM + j * 16 + l16;
      float attn = (accv[j][v] + vs[j]) * dn[v];
      float cv = 0.5f * gcn[idx] + 0.5f * attn;
      cur[idx] = cv;
      accbuf[idx] += cv;
    }
  }
}

// ---------------- launcher ----------------

extern "C" void kernel_launch(void* const* d_in, const int* in_sizes, int n_in,
                              void* d_out, int out_size, void* d_ws, size_t ws_size,
                              hipStream_t stream) {
  const float* x    = (const float*)d_in[0];
  const int*   eidx = (const int*)d_in[1];
  const float* ew   = (const float*)d_in[2];
  const int*   nn   = (const int*)d_in[3];
  const float* Wq_w = (const float*)d_in[4];
  const float* Wq_b = (const float*)d_in[5];
  const float* Wk_w = (const float*)d_in[6];
  const float* Wk_b = (const float*)d_in[7];
  const float* Wo_w = (const float*)d_in[8];
  const float* Wo_b = (const float*)d_in[9];
  float* out = (float*)d_out;

  const int N = in_sizes[0] / CDIM;
  const int E = in_sizes[1] / 2;
  const int B = in_sizes[3];
  const int* row = eidx;
  const int* col = eidx + E;
  const long long NC = (long long)N * CDIM;

  // workspace carve (256B aligned)
  char* p = (char*)d_ws;
  auto carve = [&](size_t bytes) -> void* {
    void* r = (void*)p;
    p += (bytes + 255) & ~(size_t)255;
    return r;
  };
  float* qs    = (float*)carve(sizeof(float) * NC);
  float* ks    = (float*)carve(sizeof(float) * NC);
  float* cur   = (float*)carve(sizeof(float) * NC);
  float* accb  = (float*)carve(sizeof(float) * NC);
  float* gcn   = (float*)carve(sizeof(float) * NC);
  float* S     = (float*)carve(sizeof(float) * (size_t)B * CDIM * CDIM);
  float* z     = (float*)carve(sizeof(float) * (size_t)B * CDIM);
  float* vsum  = (float*)carve(sizeof(float) * (size_t)B * CDIM);
  float* dinv  = (float*)carve(sizeof(float) * (size_t)N);
  float* denom = (float*)carve(sizeof(float) * (size_t)N);
  int*   batch = (int*)carve(sizeof(int) * (size_t)N);
  int*   offs  = (int*)carve(sizeof(int) * (size_t)(B + 1));

  const int TPB = 256;
  const size_t LDS64K = (size_t)CDIM * CDIM * sizeof(float);  // 64 KB

  // graph bookkeeping
  k_scan_offsets<<<1, 32, 0, stream>>>(nn, offs, B);
  k_assign_batch<<<(N + TPB - 1) / TPB, TPB, 0, stream>>>(offs, batch, N, B);

  // degree -> dinv
  k_zero_f32<<<1024, TPB, 0, stream>>>(dinv, N);
  k_deg<<<(E + TPB - 1) / TPB, TPB, 0, stream>>>(row, ew, dinv, E);
  k_dinv<<<(N + TPB - 1) / TPB, TPB, 0, stream>>>(dinv, N);

  // cur = acc = x
  k_init_cur_acc<<<(int)((NC + TPB - 1) / TPB), TPB, 0, stream>>>(x, cur, accb, NC);

  // Q/K projections (+ bias, row-normalize) on fp32 WMMA, W staged in LDS
  int gemm_blocks = (N / 16 + 7) / 8;  // 8 waves per 256-thread block
  k_wmma_gemm_nt<<<gemm_blocks, 256, LDS64K, stream>>>(x, Wq_w, Wq_b, qs, N, 1);
  k_wmma_gemm_nt<<<gemm_blocks, 256, LDS64K, stream>>>(x, Wk_w, Wk_b, ks, N, 1);

  for (int it = 0; it < 4; ++it) {
    // per-graph reductions and K^T V
    k_sumkv<<<B, CDIM, 0, stream>>>(ks, cur, offs, z, vsum);
    k_wmma_ktv<<<B * 8, 256, 0, stream>>>(ks, cur, offs, S);
    k_denom<<<(N + TPB - 1) / TPB, TPB, 0, stream>>>(qs, z, batch, nn, denom, N);

    // GCN scatter
    k_zero_f32<<<2048, TPB, 0, stream>>>(gcn, NC);
    long long ethreads = (long long)E * 32;
    k_edge_scatter<<<(int)((ethreads + TPB - 1) / TPB), TPB, 0, stream>>>(
        row, col, ew, dinv, cur, gcn, E);

    // attn numerator GEMM (S_b staged in LDS) + fused update of cur/acc
    k_wmma_attn_fuse<<<gemm_blocks, 256, LDS64K, stream>>>(qs, S, vsum, denom, gcn,
                                                           batch, cur, accb, N);
  }

  // output projection (bias, no normalize)
  k_wmma_gemm_nt<<<gemm_blocks, 256, LDS64K, stream>>>(accb, Wo_w, Wo_b, out, N, 0);
}